// DiffeqSolver_53584011985384
// MI455X (gfx1250) — compile-verified
//
#include <hip/hip_runtime.h>
#include <hip/hip_bf16.h>

// ---------------------------------------------------------------------------
// Neural-ODE RK4 rollout on gfx1250 using V_WMMA_F32_16X16X4_F32.
//   rows = B*N = 32768, L = 64, H = 256, T = 32 (31 RK4 steps)
// One wave integrates a 16-row tile for all 31 steps; weights live in LDS
// (transposed) for the whole kernel; only the trajectory is written to HBM.
// ---------------------------------------------------------------------------

typedef __attribute__((ext_vector_type(2))) float v2f;
typedef __attribute__((ext_vector_type(8))) float v8f;

#define NUM_T      32
#define ROWS       (32 * 1024)          // B*N
#define LAT        64                   // L
#define HID        256                  // H
#define ROWS_X_LAT (ROWS * LAT)         // 2097152
#define WAVES_WG   8
#define ROWS_WG    (16 * WAVES_WG)      // 128

#if __has_builtin(__builtin_amdgcn_tanhf)
  #define TANH(x) __builtin_amdgcn_tanhf(x)
#elif __has_builtin(__builtin_amdgcn_tanh_f32)
  #define TANH(x) __builtin_amdgcn_tanh_f32(x)
#else
  #define TANH(x) tanhf(x)
#endif

#define WMMA_F32(a, b, c) \
  __builtin_amdgcn_wmma_f32_16x16x4_f32(false, (a), false, (b), (short)0, (c), false, false)

static __device__ __forceinline__ v8f splat8(float x) {
  v8f r;
#pragma unroll
  for (int i = 0; i < 8; ++i) r[i] = x;
  return r;
}

// f(y) = tanh(y @ W1 + b1) @ W2 + b2  on a 16x64 tile owned by one wave.
// in/outk are 4 C-layout fragments (16x16 each, v8f per lane).
// sX: per-wave LDS scratch (16*64 floats) used for C->A layout conversion.
// sW1t: W1 transposed [n][k] (256 x 64); sW2t: W2 transposed [n][k] (64 x 256).
static __device__ __forceinline__ void feval(const v8f in[4], v8f outk[4],
                                             float* __restrict__ sX,
                                             const float* __restrict__ sW1t,
                                             const float* __restrict__ sW2t,
                                             const float* __restrict__ sb1,
                                             const float* __restrict__ sb2,
                                             int lm, int g) {
  // 1) Spill the input tile to LDS, row-major [16][64].
  //    C-layout frag nb, VGPR v: element (M = v + 8*g, N = nb*16 + lm).
#pragma unroll
  for (int nb = 0; nb < 4; ++nb) {
#pragma unroll
    for (int v = 0; v < 8; ++v) {
      sX[(v + 8 * g) * LAT + nb * 16 + lm] = in[nb][v];
    }
  }

  // 2) Gather A-layout fragments: for K-block kb, lane (lm,g) holds
  //    X[M=lm][K = 4*kb + 2*g + {0,1}]  (ISA 7.12.2, 32-bit A 16x4).
  v2f xa[16];
#pragma unroll
  for (int kb = 0; kb < 16; ++kb) {
    xa[kb] = *(const v2f*)&sX[lm * LAT + kb * 4 + 2 * g];
  }

  // 3) Output accumulators start from bias b2 (per-column broadcast).
#pragma unroll
  for (int nb = 0; nb < 4; ++nb) outk[nb] = splat8(sb2[nb * 16 + lm]);

  // 4) Stream the 16 hidden 16-wide blocks; fuse tanh + second GEMM.
  for (int hb = 0; hb < 16; ++hb) {
    v8f c = splat8(sb1[hb * 16 + lm]);
    // GEMM1: c += X[16x64] @ W1[:, hb*16 : hb*16+16]
#pragma unroll
    for (int kb = 0; kb < 16; ++kb) {
      // B-frag (4x16): lane holds W1[k0..k0+1][n], k0 = 4*kb + 2*g, n = hb*16+lm
      v2f bf = *(const v2f*)&sW1t[(hb * 16 + lm) * LAT + kb * 4 + 2 * g];
      c = WMMA_F32(xa[kb], bf, c);
    }
#pragma unroll
    for (int e = 0; e < 8; ++e) c[e] = TANH(c[e]);

    // C->A layout conversion of the 16x16 hidden block via LDS scratch head.
#pragma unroll
    for (int v = 0; v < 8; ++v) {
      sX[(v + 8 * g) * 16 + lm] = c[v];
    }
    v2f ha[4];
#pragma unroll
    for (int q = 0; q < 4; ++q) {
      ha[q] = *(const v2f*)&sX[lm * 16 + q * 4 + 2 * g];
    }

    // GEMM2: outk[nb] += H_block[16x16] @ W2[hb*16 : hb*16+16, nb*16 : +16]
#pragma unroll
    for (int q = 0; q < 4; ++q) {
#pragma unroll
      for (int nb = 0; nb < 4; ++nb) {
        v2f bf = *(const v2f*)&sW2t[(nb * 16 + lm) * HID + hb * 16 + q * 4 + 2 * g];
        outk[nb] = WMMA_F32(ha[q], bf, outk[nb]);
      }
    }
  }
}

__global__ __launch_bounds__(256, 1)
void node_rk4_kernel(const float* __restrict__ fp,   // [ROWS, LAT]
                     const float* __restrict__ ts,   // [NUM_T]
                     const float* __restrict__ W1,   // [LAT, HID] row-major
                     const float* __restrict__ b1,   // [HID]
                     const float* __restrict__ W2,   // [HID, LAT] row-major
                     const float* __restrict__ b2,   // [LAT]
                     float* __restrict__ out)        // [NUM_T, ROWS, LAT]
{
  extern __shared__ float smem[];
  float* sW1t  = smem;                    // 256*64   = 16384  : W1^T [n][k]
  float* sW2t  = sW1t + LAT * HID;        // 64*256   = 16384  : W2^T [n][k]
  float* sb1   = sW2t + HID * LAT;        // 256
  float* sb2   = sb1 + HID;               // 64
  float* sts   = sb2 + LAT;               // 32
  float* sXall = sts + NUM_T;             // 8 * 16*64 = 8192

  const int tid = threadIdx.x;

  // Stage weights (transposed) + biases + timeline into LDS once.
  for (int i = tid; i < LAT * HID; i += 256) {
    int k = i >> 8, n = i & 255;          // W1[k][n], k<64, n<256
    sW1t[n * LAT + k] = W1[i];
  }
  for (int i = tid; i < HID * LAT; i += 256) {
    int k = i >> 6, n = i & 63;           // W2[k][n], k<256, n<64
    sW2t[n * HID + k] = W2[i];
  }
  if (tid < HID) sb1[tid] = b1[tid];
  if (tid < LAT) sb2[tid] = b2[tid];
  if (tid < NUM_T) sts[tid] = ts[tid];
  __syncthreads();

  const int lane = tid & 31;
  const int wave = tid >> 5;
  const int lm = lane & 15;               // column / row-mod-16 index
  const int g  = lane >> 4;               // lane-half selector
  float* sX = sXall + wave * (16 * LAT);
  const int row0 = blockIdx.x * ROWS_WG + wave * 16;

  v8f y[4], ksum[4], kk[4], tmp[4];

  // Load y0 in C-layout and emit the t=0 slice.
#pragma unroll
  for (int nb = 0; nb < 4; ++nb) {
#pragma unroll
    for (int v = 0; v < 8; ++v) {
      size_t idx = (size_t)(row0 + v + 8 * g) * LAT + nb * 16 + lm;
      float val = fp[idx];
      y[nb][v] = val;
      out[idx] = val;                     // t = 0 (odeint includes t0)
    }
  }

#pragma unroll 1
  for (int step = 0; step < NUM_T - 1; ++step) {
    const float hstep = sts[step + 1] - sts[step];

    // k1
    feval(y, kk, sX, sW1t, sW2t, sb1, sb2, lm, g);
#pragma unroll
    for (int nb = 0; nb < 4; ++nb) {
      ksum[nb] = kk[nb];
      tmp[nb]  = y[nb] + (0.5f * hstep) * kk[nb];
    }
    // k2
    feval(tmp, kk, sX, sW1t, sW2t, sb1, sb2, lm, g);
#pragma unroll
    for (int nb = 0; nb < 4; ++nb) {
      ksum[nb] = ksum[nb] + 2.0f * kk[nb];
      tmp[nb]  = y[nb] + (0.5f * hstep) * kk[nb];
    }
    // k3
    feval(tmp, kk, sX, sW1t, sW2t, sb1, sb2, lm, g);
#pragma unroll
    for (int nb = 0; nb < 4; ++nb) {
      ksum[nb] = ksum[nb] + 2.0f * kk[nb];
      tmp[nb]  = y[nb] + hstep * kk[nb];
    }
    // k4
    feval(tmp, kk, sX, sW1t, sW2t, sb1, sb2, lm, g);
#pragma unroll
    for (int nb = 0; nb < 4; ++nb) {
      ksum[nb] = ksum[nb] + kk[nb];
      y[nb] = y[nb] + (hstep * (1.0f / 6.0f)) * ksum[nb];
    }

    // Emit slice t = step + 1.
    const size_t base = (size_t)(step + 1) * ROWS_X_LAT;
#pragma unroll
    for (int nb = 0; nb < 4; ++nb) {
#pragma unroll
      for (int v = 0; v < 8; ++v) {
        out[base + (size_t)(row0 + v + 8 * g) * LAT + nb * 16 + lm] = y[nb][v];
      }
    }
  }
}

extern "C" void kernel_launch(void* const* d_in, const int* in_sizes, int n_in,
                              void* d_out, int out_size, void* d_ws, size_t ws_size,
                              hipStream_t stream) {
  const float* fp = (const float*)d_in[0];  // first_point  [1,32,1024,64]
  const float* ts = (const float*)d_in[1];  // time steps   [32]
  const float* W1 = (const float*)d_in[2];  // [64,256]
  const float* b1 = (const float*)d_in[3];  // [256]
  const float* W2 = (const float*)d_in[4];  // [256,64]
  const float* b2 = (const float*)d_in[5];  // [64]
  float* out = (float*)d_out;               // [32, 32768, 64]

  const int blocks = ROWS / ROWS_WG;        // 256
  const size_t shmem = (size_t)(LAT * HID + HID * LAT + HID + LAT + NUM_T +
                                WAVES_WG * 16 * LAT) * sizeof(float);  // ~161.4 KB
  node_rk4_kernel<<<blocks, 256, shmem, stream>>>(fp, ts, W1, b1, W2, b2, out);
}